// SortPredictionByEta_26053271617811
// MI455X (gfx1250) — compile-verified
//
#include <hip/hip_runtime.h>
#include <hip/hip_bf16.h>

// Problem dims (fixed by the reference):
#define B_DIM 64
#define V_DIM 16384
#define F_DIM 32
#define SPLIT 4           // V-splits in pass 1 (grid.y)
#define P1_WAVES 32       // 1024 threads per pass-1 block
#define P1_WG (P1_WAVES / 2)  // 16 wave-groups (2 N-tiles each)

typedef __attribute__((ext_vector_type(2))) float v2f;
typedef __attribute__((ext_vector_type(8))) float v8f;

// ---------------------------------------------------------------------------
// Pass 1a: per-(batch, split) partial reduction via V_WMMA_F32_16X16X4_F32.
//   D = A x B + C with A(16x4): row0 = e[v..v+3], row1 = (e*eta)[v..v+3],
//   rows 2..15 = 0;  B(4x16) = frac[v..v+3][ftile..ftile+15].
//   Accumulator rows M=0 / M=1 hold sumE / sumEEta for 16 f-columns.
// CDNA5 f32 operand striping (ISA 7.12.2):
//   A 16x4 : lane L -> M = L&15 ; VGPR0 = K=(L<16?0:2), VGPR1 = K+1
//   B 4x16 : lane L -> N = L&15 ; VGPR0 = K=(L<16?0:2), VGPR1 = K+1
//   C/D    : VGPR r, lane L -> M = r + (L<16?0:8), N = L&15
// A element = e * w, where w = 1 (row 0), eta (row 1), 0 (rows 2..15).
// w is a branch-free cndmask so the eta load never sinks into a branch.
// ---------------------------------------------------------------------------
__global__ __launch_bounds__(1024)
void pass1_partial(const float* __restrict__ frac, const float* __restrict__ feat,
                   float* __restrict__ partial /* [B][SPLIT][64] */) {
  const int b    = blockIdx.x;
  const int s    = blockIdx.y;
  const int tid  = threadIdx.x;
  const int lane = tid & 31;
  const int wave = tid >> 5;
  const int tile = wave & 1;   // f-column tile: 0 -> f 0..15, 1 -> f 16..31
  const int wg   = wave >> 1;  // 0..15 wave-group (V partition)
  const int m    = lane & 15;  // A-matrix row / B-matrix column within tile
  const int half = lane >> 4;  // K-pair select: 0 -> K{0,1}, 1 -> K{2,3}

  const int vchunk = V_DIM / (SPLIT * P1_WG);          // 256 rows per wave
  const int v0     = (s * P1_WG + wg) * vchunk;

  const float*  fracB = frac + (size_t)b * V_DIM * F_DIM;
  const float2* featB = reinterpret_cast<const float2*>(feat + (size_t)b * V_DIM * 4);

  const int   n       = tile * 16 + m;            // f column this lane loads for B
  const bool  isEta   = (m == 1);                 // loop-invariant lane masks
  const float wConst  = (m == 0) ? 1.0f : 0.0f;   // weight when not the eta row

  v8f c = {};
  for (int vi = 0; vi < vchunk; vi += 4) {
    const int v  = v0 + vi;
    const int vk = v + half * 2;             // K rows handled by this lane
    // (e, eta) pairs: feat row stride = 4 floats -> float2 index vk*2.
    float2 f0 = featB[(size_t)vk * 2];       // row vk   : b64 load
    float2 f1 = featB[(size_t)vk * 2 + 2];   // row vk+1 : b64 load
    v2f a;
    a.x = f0.x * (isEta ? f0.y : wConst);    // e*1 | e*eta | e*0
    a.y = f1.x * (isEta ? f1.y : wConst);
    // B operand: frac[vk][n], frac[vk+1][n] (two coalesced 64B segments/instr)
    v2f bm;
    bm.x = fracB[(size_t)vk * F_DIM + n];
    bm.y = fracB[(size_t)(vk + 1) * F_DIM + n];
    c = __builtin_amdgcn_wmma_f32_16x16x4_f32(
        /*neg_a=*/false, a, /*neg_b=*/false, bm,
        /*c_mod=*/(short)0, c, /*reuse_a=*/false, /*reuse_b=*/false);
  }

  // c[0] lanes 0-15 = sumE (M=0), c[1] lanes 0-15 = sumEEta (M=1).
  __shared__ float sE[P1_WG][F_DIM];
  __shared__ float sEE[P1_WG][F_DIM];
  if (lane < 16) {
    sE[wg][tile * 16 + lane]  = c[0];
    sEE[wg][tile * 16 + lane] = c[1];
  }
  __syncthreads();
  // Fixed-order cross-wave reduction (deterministic).
  if (tid < F_DIM) {
    float accE = 0.0f, accEE = 0.0f;
    for (int g = 0; g < P1_WG; ++g) { accE += sE[g][tid]; accEE += sEE[g][tid]; }
    float* p = partial + ((size_t)b * SPLIT + s) * 64;
    p[tid]      = accE;
    p[32 + tid] = accEE;
  }
}

// ---------------------------------------------------------------------------
// Pass 1b: reduce splits, weighted eta + threshold, stable ascending argsort
// of the 32 values per batch (one lane per f, rank via 32 shuffles), emit
// perm[b][j] = original column index of the j-th smallest weta.
// ---------------------------------------------------------------------------
__global__ __launch_bounds__(32)
void pass1_finalize(const float* __restrict__ partial, int* __restrict__ perm) {
  const int b = blockIdx.x;
  const int f = threadIdx.x;  // 0..31, wave32: one lane per f
  float sumE = 0.0f, sumEE = 0.0f;
  for (int s = 0; s < SPLIT; ++s) {
    const float* p = partial + ((size_t)b * SPLIT + s) * 64;
    sumE  += p[f];
    sumEE += p[32 + f];
  }
  float w = sumEE / (sumE + 1e-7f);
  w = (fabsf(w) > 0.1f) ? w : 500.0f;

  // Stable rank: #{ j : w_j < w_f  or (w_j == w_f and j < f) }
  int rank = 0;
  for (int j = 0; j < 32; ++j) {
    float wj = __shfl(w, j, 32);
    rank += ((wj < w) || (wj == w && j < f)) ? 1 : 0;
  }
  __shared__ int sp[32];
  sp[rank] = f;          // perm[rank] = original index
  __syncthreads();
  perm[b * 32 + f] = sp[f];
}

// ---------------------------------------------------------------------------
// Pass 3: out[b, v, j] = frac[b, v, perm[b][j]].
// Lane = output column. Load row coalesced, permute columns in registers via
// ds_bpermute (__shfl), store coalesced. Non-temporal on this pass so the
// pass-1 copy of frac stays resident in the 192MB L2 and out doesn't pollute.
// ---------------------------------------------------------------------------
#define P3_WAVES 8
#define P3_ROWS  16
__global__ __launch_bounds__(P3_WAVES * 32)
void pass3_permute(const float* __restrict__ frac, const int* __restrict__ perm,
                   float* __restrict__ out) {
  const int b    = blockIdx.y;
  const int lane = threadIdx.x & 31;
  const int wave = threadIdx.x >> 5;
  const int p    = perm[b * 32 + lane];  // source column for output column `lane`
  const float* src = frac + (size_t)b * V_DIM * F_DIM;
  float*       dst = out  + (size_t)b * V_DIM * F_DIM;
  const int row0 = (blockIdx.x * P3_WAVES + wave) * P3_ROWS;
#pragma unroll 4
  for (int r = 0; r < P3_ROWS; ++r) {
    const size_t off = (size_t)(row0 + r) * F_DIM + lane;
    float x = __builtin_nontemporal_load(src + off);  // last use of frac
    float y = __shfl(x, p, 32);                       // out col lane = in col p
    __builtin_nontemporal_store(y, dst + off);        // streamed output
  }
}

extern "C" void kernel_launch(void* const* d_in, const int* in_sizes, int n_in,
                              void* d_out, int out_size, void* d_ws, size_t ws_size,
                              hipStream_t stream) {
  (void)in_sizes; (void)n_in; (void)out_size; (void)ws_size;
  const float* frac = (const float*)d_in[0];  // (64, 16384, 32) f32
  const float* feat = (const float*)d_in[1];  // (64, 16384, 4)  f32
  float* out = (float*)d_out;                 // (64, 16384, 32) f32

  // Workspace layout: partials [B][SPLIT][64] f32, then perm [B][32] i32.
  float* partial = (float*)d_ws;
  int*   perm    = (int*)((float*)d_ws + (size_t)B_DIM * SPLIT * 64);

  pass1_partial<<<dim3(B_DIM, SPLIT), P1_WAVES * 32, 0, stream>>>(frac, feat, partial);
  pass1_finalize<<<B_DIM, 32, 0, stream>>>(partial, perm);
  pass3_permute<<<dim3(V_DIM / (P3_WAVES * P3_ROWS), B_DIM), P3_WAVES * 32, 0, stream>>>(
      frac, perm, out);
}